// GCN_8718783611358
// MI455X (gfx1250) — compile-verified
//
#include <hip/hip_runtime.h>
#include <math.h>

typedef float v2f __attribute__((ext_vector_type(2)));
typedef float v8f __attribute__((ext_vector_type(8)));

// ---------------------------------------------------------------- utilities

__global__ void k_init_deg(float* __restrict__ deg, int n) {
    int i = blockIdx.x * blockDim.x + threadIdx.x;
    if (i < n) deg[i] = 1.0f;  // self-loop contributes 1 to every node's degree
}

__global__ void k_zero(float* __restrict__ p, int count) {
    int i = blockIdx.x * blockDim.x + threadIdx.x;
    if (i < count) p[i] = 0.0f;
}

__global__ void k_deg(const int* __restrict__ dst, float* __restrict__ deg, int E) {
    int e = blockIdx.x * blockDim.x + threadIdx.x;
    if (e < E) atomicAdd(&deg[dst[e]], 1.0f);
}

__global__ void k_rsqrt_inplace(float* __restrict__ deg, int n) {
    int i = blockIdx.x * blockDim.x + threadIdx.x;
    if (i < n) {
        float d = deg[i];
        deg[i] = (d > 0.0f) ? rsqrtf(d) : 0.0f;   // deg -> dinv
    }
}

// ---------------------------------------------------------------- GEMM 1
// h1[n,8] = x[n,256] @ W1[256,8], via V_WMMA_F32_16X16X4_F32.
// W1 staged zero-padded to [256,16] in LDS: branch-free ds B-fragment loads.
// fp32 16x4 A layout: lane L: m = L&15, koff = 2*(L>>4); VGPR0=K(koff), VGPR1=K(koff+1)
// fp32 4x16 B layout: lane L: n = L&15, same koff pairing.
// fp32 16x16 C/D:     VGPR r: row = r + 8*(L>>4), col = L&15.
__global__ void k_gemm1_wmma(const float* __restrict__ x,
                             const float* __restrict__ W1,
                             float* __restrict__ h1, int n) {
    __shared__ float Ws[256 * 16];           // 16 KB padded W1
    for (int idx = threadIdx.x; idx < 256 * 16; idx += blockDim.x) {
        int r = idx >> 4, cidx = idx & 15;
        Ws[idx] = (cidx < 8) ? W1[(size_t)r * 8 + cidx] : 0.0f;
    }
    __syncthreads();                         // before any wave can exit

    int wave  = (blockIdx.x * blockDim.x + threadIdx.x) >> 5;
    int lane  = threadIdx.x & 31;
    int tiles = (n + 15) >> 4;
    if (wave >= tiles) return;               // wave-uniform: WMMA waves keep full EXEC

    int half = lane >> 4;                    // 0 | 1
    int l15  = lane & 15;
    int koff = half * 2;

    int row = wave * 16 + l15;
    if (row >= n) row = n - 1;               // clamp (store is guarded on tail tiles)
    const float* xrow = x + (size_t)row * 256;

    v8f c = {0.f, 0.f, 0.f, 0.f, 0.f, 0.f, 0.f, 0.f};

    #pragma unroll 4
    for (int k = 0; k < 256; k += 4) {
        v2f a, b;
        a.x = xrow[k + koff];
        a.y = xrow[k + koff + 1];
        b.x = Ws[(k + koff)     * 16 + l15];
        b.y = Ws[(k + koff + 1) * 16 + l15];
        c = __builtin_amdgcn_wmma_f32_16x16x4_f32(false, a, false, b,
                                                  (short)0, c, false, false);
    }

    // epilogue: fast path for full tiles (wave-uniform), guarded path for tail
    float* orow0 = h1 + (size_t)(wave * 16 + half * 8) * 8 + l15;
    if (wave * 16 + 16 <= n) {
        if (l15 < 8) {
            #pragma unroll
            for (int r = 0; r < 8; ++r) orow0[(size_t)r * 8] = c[r];
        }
    } else {
        if (l15 < 8) {
            #pragma unroll
            for (int r = 0; r < 8; ++r) {
                int orow = wave * 16 + r + half * 8;
                if (orow < n) h1[(size_t)orow * 8 + l15] = c[r];
            }
        }
    }
}

// ---------------------------------------------------------------- GEMM 2
// h2[n,16] = hrelu[n,8] @ W2[8,16]; K=8 -> two WMMA K-steps, N=16 exact.
__global__ void k_gemm2_wmma(const float* __restrict__ hin,
                             const float* __restrict__ W2,
                             float* __restrict__ h2, int n) {
    __shared__ float Ws[8 * 16];
    if (threadIdx.x < 128) Ws[threadIdx.x] = W2[threadIdx.x];
    __syncthreads();

    int wave  = (blockIdx.x * blockDim.x + threadIdx.x) >> 5;
    int lane  = threadIdx.x & 31;
    int tiles = (n + 15) >> 4;
    if (wave >= tiles) return;

    int half = lane >> 4;
    int l15  = lane & 15;
    int koff = half * 2;

    int row = wave * 16 + l15;
    if (row >= n) row = n - 1;
    const float* hrow = hin + (size_t)row * 8;

    v8f c = {0.f, 0.f, 0.f, 0.f, 0.f, 0.f, 0.f, 0.f};
    #pragma unroll
    for (int k = 0; k < 8; k += 4) {
        v2f a, b;
        a.x = hrow[k + koff];
        a.y = hrow[k + koff + 1];
        b.x = Ws[(k + koff)     * 16 + l15];
        b.y = Ws[(k + koff + 1) * 16 + l15];
        c = __builtin_amdgcn_wmma_f32_16x16x4_f32(false, a, false, b,
                                                  (short)0, c, false, false);
    }

    float* orow0 = h2 + (size_t)(wave * 16 + half * 8) * 16 + l15;
    if (wave * 16 + 16 <= n) {
        #pragma unroll
        for (int r = 0; r < 8; ++r) orow0[(size_t)r * 16] = c[r];
    } else {
        #pragma unroll
        for (int r = 0; r < 8; ++r) {
            int orow = wave * 16 + r + half * 8;
            if (orow < n) h2[(size_t)orow * 16 + l15] = c[r];
        }
    }
}

// ---------------------------------------------------------------- scatter
// One thread per (edge, column): coalesced gather of h[s*W + j], f32 atomics out.
__global__ void k_scatter8(const int* __restrict__ src, const int* __restrict__ dst,
                           const float* __restrict__ dinv, const float* __restrict__ h,
                           float* __restrict__ agg, int total) {
    int tid = blockIdx.x * blockDim.x + threadIdx.x;
    if (tid >= total) return;
    int e = tid >> 3, j = tid & 7;
    int s = src[e], d = dst[e];
    float nrm = dinv[s] * dinv[d];
    atomicAdd(&agg[(size_t)d * 8 + j], nrm * h[(size_t)s * 8 + j]);
}

__global__ void k_scatter16(const int* __restrict__ src, const int* __restrict__ dst,
                            const float* __restrict__ dinv, const float* __restrict__ h,
                            float* __restrict__ agg, int total) {
    int tid = blockIdx.x * blockDim.x + threadIdx.x;
    if (tid >= total) return;
    int e = tid >> 4, j = tid & 15;
    int s = src[e], d = dst[e];
    float nrm = dinv[s] * dinv[d];
    atomicAdd(&agg[(size_t)d * 16 + j], nrm * h[(size_t)s * 16 + j]);
}

// ---------------------------------------------------------------- finalize
// layer1: agg += dinv^2 * h (self-loop) + b1, then ReLU, in-place in agg1.
__global__ void k_fin1(float* __restrict__ agg, const float* __restrict__ h,
                       const float* __restrict__ dinv, const float* __restrict__ b1,
                       int total) {
    int tid = blockIdx.x * blockDim.x + threadIdx.x;
    if (tid >= total) return;
    int i = tid >> 3, j = tid & 7;
    float di = dinv[i];
    float v = agg[tid] + di * di * h[tid] + b1[j];
    agg[tid] = v > 0.0f ? v : 0.0f;
}

// layer2: add self-loop + bias, then row-wise log_softmax over 16 classes.
// float4 row traffic -> global_load_b128 / global_store_b128.
__global__ void k_fin2(const float* __restrict__ agg, const float* __restrict__ h,
                       const float* __restrict__ dinv, const float* __restrict__ b2,
                       float* __restrict__ out, int n) {
    int i = blockIdx.x * blockDim.x + threadIdx.x;
    if (i >= n) return;
    float di = dinv[i];
    float d2 = di * di;

    const float4* ag4 = (const float4*)(agg + (size_t)i * 16);
    const float4* h4  = (const float4*)(h   + (size_t)i * 16);
    const float4* b4  = (const float4*)b2;

    float v[16];
    float mx = -INFINITY;
    #pragma unroll
    for (int q = 0; q < 4; ++q) {
        float4 a = ag4[q], hh = h4[q], bb = b4[q];
        v[4*q + 0] = a.x + d2 * hh.x + bb.x;
        v[4*q + 1] = a.y + d2 * hh.y + bb.y;
        v[4*q + 2] = a.z + d2 * hh.z + bb.z;
        v[4*q + 3] = a.w + d2 * hh.w + bb.w;
    }
    #pragma unroll
    for (int j = 0; j < 16; ++j) mx = fmaxf(mx, v[j]);
    float s = 0.0f;
    #pragma unroll
    for (int j = 0; j < 16; ++j) s += __expf(v[j] - mx);
    float lse = __logf(s) + mx;

    float4* o4 = (float4*)(out + (size_t)i * 16);
    #pragma unroll
    for (int q = 0; q < 4; ++q) {
        float4 r;
        r.x = v[4*q + 0] - lse;
        r.y = v[4*q + 1] - lse;
        r.z = v[4*q + 2] - lse;
        r.w = v[4*q + 3] - lse;
        o4[q] = r;
    }
}

// ---------------------------------------------------------------- launch

extern "C" void kernel_launch(void* const* d_in, const int* in_sizes, int n_in,
                              void* d_out, int out_size, void* d_ws, size_t ws_size,
                              hipStream_t stream) {
    const float* x   = (const float*)d_in[0];
    const int*  edge = (const int*) d_in[1];
    const float* W1  = (const float*)d_in[2];
    const float* b1  = (const float*)d_in[3];
    const float* W2  = (const float*)d_in[4];
    const float* b2  = (const float*)d_in[5];
    float* out = (float*)d_out;

    const int n = in_sizes[0] / 256;   // D_IN = 256
    const int E = in_sizes[1] / 2;
    const int* src = edge;
    const int* dst = edge + E;

    // workspace: deg/dinv(n) | h1(8n) | h2(16n) | agg1(8n) | agg2(16n)  = 49n floats
    float* w    = (float*)d_ws;
    float* deg  = w;                         // becomes dinv in place
    float* h1   = deg  + n;
    float* h2   = h1   + (size_t)8  * n;
    float* agg1 = h2   + (size_t)16 * n;
    float* agg2 = agg1 + (size_t)8  * n;     // contiguous with agg1

    const int B = 256;
    auto nb = [](long total, int b) { return (int)((total + b - 1) / b); };

    // degrees + normalization
    k_init_deg<<<nb(n, B), B, 0, stream>>>(deg, n);
    k_zero<<<nb((long)24 * n, B), B, 0, stream>>>(agg1, 24 * n);  // agg1+agg2
    k_deg<<<nb(E, B), B, 0, stream>>>(dst, deg, E);
    k_rsqrt_inplace<<<nb(n, B), B, 0, stream>>>(deg, n);

    // layer 1
    const int tiles   = (n + 15) / 16;
    const int gblocks = (tiles + 7) / 8;     // 8 waves per 256-thread block
    k_gemm1_wmma<<<gblocks, B, 0, stream>>>(x, W1, h1, n);
    k_scatter8<<<nb((long)E * 8, B), B, 0, stream>>>(src, dst, deg, h1, agg1, E * 8);
    k_fin1<<<nb((long)n * 8, B), B, 0, stream>>>(agg1, h1, deg, b1, n * 8);

    // layer 2
    k_gemm2_wmma<<<gblocks, B, 0, stream>>>(agg1, W2, h2, n);
    k_scatter16<<<nb((long)E * 16, B), B, 0, stream>>>(src, dst, deg, h2, agg2, E * 16);
    k_fin2<<<nb(n, B), B, 0, stream>>>(agg2, h2, deg, b2, out, n);
}